// VRGCNConv_34394098106414
// MI455X (gfx1250) — compile-verified
//
#include <hip/hip_runtime.h>

// VRGCN forward for MI455X (gfx1250, wave32).
// ws layout (floats): [0..127] sum(x), [128..255] sum(x^2),
//                     [256..383] mean, [384..511] rstd
#define DIM 128

typedef __attribute__((ext_vector_type(2))) float v2f;
typedef __attribute__((ext_vector_type(8))) float v8f;

// ---------------------------------------------------------------------------
// Phase 1: per-feature column sums of x and x^2 via V_WMMA_F32_16X16X4_F32.
// A = 16x4 ones; B = 4 rows x 16 features of x. Every row of D = column sums.
// One wave (32 threads) per block; grid = 8 feature-chunks * nslice row-slices.
// e is a multiple of 4 (50000), so the K loop is exact.
// ---------------------------------------------------------------------------
__global__ __launch_bounds__(32) void vrgcn_stats_wmma(
    const float* __restrict__ x, float* __restrict__ ws, int e, int nslice) {
  const int chunk = blockIdx.x & 7;   // which 16-feature column chunk
  const int slice = blockIdx.x >> 3;  // which row slice
  const int lane  = threadIdx.x;      // 0..31 (wave32)
  const int col   = chunk * 16 + (lane & 15);
  const int rbase = (lane < 16) ? 0 : 2;  // K-layout: lanes 0-15 hold K=0,1; 16-31 hold K=2,3

  v2f a; a[0] = 1.0f; a[1] = 1.0f;        // 16x4 all-ones A matrix
  v8f acc_s = {};                          // accumulates sum(x)
  v8f acc_q = {};                          // accumulates sum(x^2)

  const int ksteps = e >> 2;               // groups of 4 rows
  for (int k = slice; k < ksteps; k += nslice) {
    const size_t base = (size_t)(k * 4 + rbase) * DIM + col;
    const float b0 = x[base];
    const float b1 = x[base + DIM];
    v2f b;  b[0]  = b0;      b[1]  = b1;
    v2f bq; bq[0] = b0 * b0; bq[1] = b1 * b1;
    acc_s = __builtin_amdgcn_wmma_f32_16x16x4_f32(
        false, a, false, b,  (short)0, acc_s, false, false);
    acc_q = __builtin_amdgcn_wmma_f32_16x16x4_f32(
        false, a, false, bq, (short)0, acc_q, false, false);
  }

  // D VGPR0, lanes 0..15 = row M=0 => the 16 column sums of this chunk.
  if (lane < 16) {
    atomicAdd(&ws[col], acc_s[0]);
    atomicAdd(&ws[DIM + col], acc_q[0]);
  }
}

// ---------------------------------------------------------------------------
// Phase 2: mean / rstd (biased variance, BN eps = 1e-5), 128 threads.
// ---------------------------------------------------------------------------
__global__ void vrgcn_finalize_stats(float* __restrict__ ws, float inv_e) {
  const int c = threadIdx.x;
  const float mean = ws[c] * inv_e;
  const float var  = ws[DIM + c] * inv_e - mean * mean;
  ws[2 * DIM + c] = mean;
  ws[3 * DIM + c] = __frsqrt_rn(var + 1e-5f);
}

// ---------------------------------------------------------------------------
// Phase 3: out = normalized(x) * kernels   (the "+ x" term of the result).
// float4 vectorized over the 6.4M elements.
// ---------------------------------------------------------------------------
__global__ __launch_bounds__(256) void vrgcn_norm_init(
    const float* __restrict__ x, const float* __restrict__ ws,
    const float* __restrict__ gamma, const float* __restrict__ beta,
    const float* __restrict__ kernels, float* __restrict__ out, int n4) {
  const int i = blockIdx.x * blockDim.x + threadIdx.x;
  if (i >= n4) return;
  const int c = (i * 4) & (DIM - 1);
  const float4 xv = ((const float4*)x)[i];
  const float4 mu = *(const float4*)(ws + 2 * DIM + c);
  const float4 rs = *(const float4*)(ws + 3 * DIM + c);
  const float4 g  = *(const float4*)(gamma + c);
  const float4 bb = *(const float4*)(beta + c);
  const float4 kk = *(const float4*)(kernels + c);
  float4 o;
  o.x = ((xv.x - mu.x) * rs.x * g.x + bb.x) * kk.x;
  o.y = ((xv.y - mu.y) * rs.y * g.y + bb.y) * kk.y;
  o.z = ((xv.z - mu.z) * rs.z * g.z + bb.z) * kk.z;
  o.w = ((xv.w - mu.w) * rs.w * g.w + bb.w) * kk.w;
  ((float4*)out)[i] = o;
}

// ---------------------------------------------------------------------------
// Phase 4: edge scatter. One wave per edge (lane*4 = 4 contiguous features),
// grid-stride over edges so the 128-wide BN/affine params are loaded once per
// wave. Normalization of gathered rows is recomputed in registers; the
// scatter-adds are GLOBAL_ATOMIC_ADD_F32 that land in the 192MB L2
// (x and out are both fully L2-resident).
// ---------------------------------------------------------------------------
__global__ __launch_bounds__(256) void vrgcn_edge_scatter(
    const float* __restrict__ x, const float* __restrict__ r,
    const float* __restrict__ ws,
    const float* __restrict__ gamma, const float* __restrict__ beta,
    const float* __restrict__ kernels,
    const int* __restrict__ heads, const int* __restrict__ tails,
    const int* __restrict__ rels,
    float* __restrict__ out, int E) {
  const int lane   = threadIdx.x & 31;
  const int wave   = (int)((blockIdx.x * blockDim.x + threadIdx.x) >> 5);
  const int nwaves = (int)((gridDim.x * blockDim.x) >> 5);
  const int c = lane * 4;

  const float4 mu = *(const float4*)(ws + 2 * DIM + c);
  const float4 rs = *(const float4*)(ws + 3 * DIM + c);
  const float4 g  = *(const float4*)(gamma + c);
  const float4 bb = *(const float4*)(beta + c);
  const float4 kk = *(const float4*)(kernels + c);

  for (int eid = wave; eid < E; eid += nwaves) {
    const int h  = heads[eid];
    const int t  = tails[eid];
    const int rl = rels[eid];
    const float4 xh = *(const float4*)(x + (size_t)h * DIM + c);
    const float4 xt = *(const float4*)(x + (size_t)t * DIM + c);
    const float4 rr = *(const float4*)(r + (size_t)rl * DIM + c);
    float* __restrict__ oh = out + (size_t)h * DIM + c;
    float* __restrict__ ot = out + (size_t)t * DIM + c;

    float nh, nt, rk;
    nh = ((xh.x - mu.x) * rs.x * g.x + bb.x) * kk.x;
    nt = ((xt.x - mu.x) * rs.x * g.x + bb.x) * kk.x;
    rk = rr.x * kk.x;
    atomicAdd(ot + 0, nh + rk);
    atomicAdd(oh + 0, nt - rk);

    nh = ((xh.y - mu.y) * rs.y * g.y + bb.y) * kk.y;
    nt = ((xt.y - mu.y) * rs.y * g.y + bb.y) * kk.y;
    rk = rr.y * kk.y;
    atomicAdd(ot + 1, nh + rk);
    atomicAdd(oh + 1, nt - rk);

    nh = ((xh.z - mu.z) * rs.z * g.z + bb.z) * kk.z;
    nt = ((xt.z - mu.z) * rs.z * g.z + bb.z) * kk.z;
    rk = rr.z * kk.z;
    atomicAdd(ot + 2, nh + rk);
    atomicAdd(oh + 2, nt - rk);

    nh = ((xh.w - mu.w) * rs.w * g.w + bb.w) * kk.w;
    nt = ((xt.w - mu.w) * rs.w * g.w + bb.w) * kk.w;
    rk = rr.w * kk.w;
    atomicAdd(ot + 3, nh + rk);
    atomicAdd(oh + 3, nt - rk);
  }
}

// ---------------------------------------------------------------------------
// Phase 5: divide by the quirky degree array. du == 1 everywhere except up to
// six scalar fix-ups derived from element 0 vs element 2 of heads/rels/tails,
// exactly replicating the reference's scalar-indexed .at[].add().
// ---------------------------------------------------------------------------
__global__ __launch_bounds__(256) void vrgcn_degree_div(
    float* __restrict__ out,
    const int* __restrict__ heads, const int* __restrict__ tails,
    const int* __restrict__ rels, int n) {
  const int i = blockIdx.x * blockDim.x + threadIdx.x;
  if (i >= n) return;
  const int node = i >> 7;  // DIM == 128
  const int h0 = heads[0], h2 = heads[2];
  const int r0 = rels[0],  r2 = rels[2];
  const int t0 = tails[0], t2 = tails[2];
  float du = 1.0f;
  if (h0 != h2) du += (float)((node == h0) + (node == h2));
  if (r0 != r2) du += (float)((node == r0) + (node == r2));
  if (t0 != t2) du += (float)((node == t0) + (node == t2));
  out[i] /= du;
}

// ---------------------------------------------------------------------------
extern "C" void kernel_launch(void* const* d_in, const int* in_sizes, int n_in,
                              void* d_out, int out_size, void* d_ws, size_t ws_size,
                              hipStream_t stream) {
  const float* x       = (const float*)d_in[0];
  const float* r       = (const float*)d_in[1];
  const float* gamma   = (const float*)d_in[2];
  const float* beta    = (const float*)d_in[3];
  const float* kernels = (const float*)d_in[4];
  const int*   edges   = (const int*)d_in[5];
  const int*   rels    = (const int*)d_in[6];
  float* out = (float*)d_out;
  float* ws  = (float*)d_ws;

  const int e = in_sizes[0] / DIM;   // 50000 nodes
  const int E = in_sizes[6];         // 625000 edges
  const int* heads = edges;
  const int* tails = edges + E;

  // zero the sum/sumsq accumulators (graph-capture-safe)
  hipMemsetAsync(d_ws, 0, 2 * DIM * sizeof(float), stream);

  const int nslice = 128;  // row slices per feature chunk
  vrgcn_stats_wmma<<<dim3(8 * nslice), dim3(32), 0, stream>>>(x, ws, e, nslice);
  vrgcn_finalize_stats<<<1, DIM, 0, stream>>>(ws, 1.0f / (float)e);

  const int n  = e * DIM;
  const int n4 = n / 4;
  vrgcn_norm_init<<<(n4 + 255) / 256, 256, 0, stream>>>(x, ws, gamma, beta,
                                                        kernels, out, n4);
  vrgcn_edge_scatter<<<2048, 256, 0, stream>>>(x, r, ws, gamma, beta, kernels,
                                               heads, tails, rels, out, E);
  vrgcn_degree_div<<<(n + 255) / 256, 256, 0, stream>>>(out, heads, tails,
                                                        rels, n);
}